// PM25GNN_16509854286702
// MI455X (gfx1250) — compile-verified
//
#include <hip/hip_runtime.h>
#include <hip/hip_bf16.h>
#include <math.h>

#define B_     4
#define N_     10000
#define E_     160000
#define HIST_  8
#define PRED_  12
#define SEQ_   20          // HIST + PRED
#define F_     8           // exogenous features
#define IN_    9           // IN_DIM
#define HID_   64
#define GOUT_  13          // GNN_OUT
#define EH_    32          // E_HID
#define EO_    30          // E_OUT
#define XC_    22          // GRU_IN = IN_DIM + GNN_OUT
#define G3_    192         // 3*HID

typedef __attribute__((ext_vector_type(2))) float v2f;
typedef __attribute__((ext_vector_type(8))) float v8f;
typedef __attribute__((ext_vector_type(4))) unsigned u32x4;
typedef __attribute__((ext_vector_type(8))) unsigned u32x8;

__device__ __forceinline__ v8f wmma4(v2f a, v2f b, v8f c) {
  // D = A(16x4 f32) * B(4x16 f32) + C(16x16 f32)
  return __builtin_amdgcn_wmma_f32_16x16x4_f32(false, a, false, b, (short)0, c,
                                               false, false);
}
__device__ __forceinline__ float sigf(float x) { return 1.0f / (1.0f + __expf(-x)); }

// 1D TDM copy: nelem dwords from global -> LDS (contiguous), tracked by TENSORcnt.
// D# built per CDNA5 ISA (count=1, data_size=4B,
// tile = 1 row of nelem elements, type=2).
__device__ __forceinline__ void tdm_load_1d(unsigned lds_off, const void* gptr,
                                            unsigned nelem) {
  unsigned long long ga = (unsigned long long)(uintptr_t)gptr;
  u32x4 g0;
  g0[0] = 1u;                                          // count=1, user mode
  g0[1] = lds_off;                                     // lds_addr (bytes)
  g0[2] = (unsigned)(ga & 0xFFFFFFFFu);                // global_addr[31:0]
  g0[3] = (unsigned)((ga >> 32) & 0x01FFFFFFu) | (2u << 30);  // addr[56:32]|type=2
  u32x8 g1;
  g1[0] = (2u << 16);                                  // workgroup_mask=0, data_size=4B
  g1[1] = (nelem & 0xFFFFu) << 16;                     // tensor_dim0[15:0]
  g1[2] = (nelem >> 16) | (1u << 16);                  // tensor_dim0[31:16], tensor_dim1=1
  g1[3] = (nelem & 0xFFFFu) << 16;                     // tile_dim0 = nelem
  g1[4] = 1u;                                          // tile_dim1 = 1
  g1[5] = nelem;                                       // tensor_dim0_stride[31:0]
  g1[6] = 0u;
  g1[7] = 0u;
  asm volatile("tensor_load_to_lds %0, %1" ::"s"(g0), "s"(g1) : "memory");
}

// ---------------------------------------------------------------- init
__global__ void k_init(const float* __restrict__ pm25, float* __restrict__ hn0,
                       float* __restrict__ xn) {
  int tid = blockIdx.x * blockDim.x + threadIdx.x;
  if (tid < B_ * N_ * HID_) hn0[tid] = 0.0f;
  if (tid < B_ * N_) {
    int b = tid / N_, n = tid % N_;
    xn[tid] = pm25[((size_t)b * HIST_ + (HIST_ - 1)) * N_ + n];  // pm25[b,HIST-1,n,0]
  }
}

__global__ void k_stats_zero(float* __restrict__ st) {
  if (threadIdx.x < 4) st[threadIdx.x] = 0.0f;
}

__global__ void k_stats(const float* __restrict__ ea, float* __restrict__ st) {
  int gid = blockIdx.x * blockDim.x + threadIdx.x;
  int stride = gridDim.x * blockDim.x;
  float s0 = 0.f, s1 = 0.f, q0 = 0.f, q1 = 0.f;
  for (int e = gid; e < E_; e += stride) {
    float a0 = ea[2 * e], a1 = ea[2 * e + 1];
    s0 += a0; s1 += a1; q0 += a0 * a0; q1 += a1 * a1;
  }
  atomicAdd(&st[0], s0); atomicAdd(&st[1], s1);
  atomicAdd(&st[2], q0); atomicAdd(&st[3], q1);
}

// per-edge time-invariant constants: ean0, ean1, direc, 3/max(dist,1e-3)
__global__ void k_edge_pre(const float* __restrict__ ea, const float* __restrict__ st,
                           float* __restrict__ ec) {
  int e = blockIdx.x * blockDim.x + threadIdx.x;
  if (e >= E_) return;
  float m0 = st[0] / (float)E_, m1 = st[1] / (float)E_;
  float v0 = (st[2] - st[0] * m0) / (float)(E_ - 1);
  float v1 = (st[3] - st[1] * m1) / (float)(E_ - 1);
  float sd0 = fmaxf(sqrtf(fmaxf(v0, 0.f)), 1e-6f);
  float sd1 = fmaxf(sqrtf(fmaxf(v1, 0.f)), 1e-6f);
  float a0 = ea[2 * e], a1 = ea[2 * e + 1];
  ec[4 * e + 0] = (a0 - m0) / sd0;
  ec[4 * e + 1] = (a1 - m1) / sd1;
  ec[4 * e + 2] = a1;                       // city_direc
  ec[4 * e + 3] = 3.0f / fmaxf(a0, 1e-3f);  // 3 / city_dist
}

// ---------------- one-time: zero-padded transposed weights (B-matrix layouts)
// wiT (24 x 192): wiT[k][c] = gwi[c][k] (k<22 else 0)
// whT (64 x 192): whT[k][c] = gwh[c][k]
// e1p (24 x 32) : e1p[k][c] = ew1[k][c] (k<21 else 0)
// e2p (32 x 32) : e2p[k][c] = ew2[k][c] (c<30 else 0)
// eb2p(32)      : padded bias
__global__ void k_pack(const float* __restrict__ gwi, const float* __restrict__ gwh,
                       const float* __restrict__ ew1, const float* __restrict__ ew2,
                       const float* __restrict__ eb2, float* __restrict__ wiT,
                       float* __restrict__ whT, float* __restrict__ e1p,
                       float* __restrict__ e2p, float* __restrict__ eb2p) {
  int tid = blockIdx.x * blockDim.x + threadIdx.x;
  int stride = gridDim.x * blockDim.x;
  for (int i = tid; i < 24 * G3_; i += stride) {
    int k = i / G3_, c = i % G3_;
    wiT[i] = (k < XC_) ? gwi[(size_t)c * XC_ + k] : 0.0f;
  }
  for (int i = tid; i < HID_ * G3_; i += stride) {
    int k = i / G3_, c = i % G3_;
    whT[i] = gwh[(size_t)c * HID_ + k];
  }
  for (int i = tid; i < 24 * EH_; i += stride) {
    int k = i / EH_, c = i % EH_;
    e1p[i] = (k < 21) ? ew1[k * EH_ + c] : 0.0f;
  }
  for (int i = tid; i < 32 * 32; i += stride) {
    int k = i / 32, c = i % 32;
    e2p[i] = (c < EO_) ? ew2[k * EO_ + c] : 0.0f;
  }
  if (tid < 32) eb2p[tid] = (tid < EO_) ? eb2[tid] : 0.0f;
}

// -------------------------------------------------- per-step: zero agg + build x
__global__ void k_step_begin(float* __restrict__ agg, float* __restrict__ x,
                             const float* __restrict__ xn,
                             const float* __restrict__ feature, int t) {
  int tid = blockIdx.x * blockDim.x + threadIdx.x;
  if (tid < B_ * N_ * EO_) agg[tid] = 0.0f;
  if (tid < B_ * N_) {
    int b = tid / N_, n = tid % N_;
    float* xr = x + (size_t)tid * IN_;
    xr[0] = xn[tid];
    const float* f = feature + (((size_t)b * SEQ_ + (HIST_ + t)) * N_ + n) * F_;
#pragma unroll
    for (int k = 0; k < F_; ++k) xr[1 + k] = f[k];
  }
}

// ---------------------------------------------------------------- edge kernel
// one wave per 16-edge tile: gather -> WMMA MLP (21->32->30, sigmoid) -> atomic scatter
#define EW_ 8  // waves per block
__global__ __launch_bounds__(256) void k_edge(
    const float* __restrict__ x, const int* __restrict__ eidx,
    const float* __restrict__ ec, const float* __restrict__ e1p,
    const float* __restrict__ eb1, const float* __restrict__ e2p,
    const float* __restrict__ eb2p, const float* __restrict__ wmean,
    const float* __restrict__ wstd, float* __restrict__ agg) {
  __shared__ float lds_in[EW_][16][24];
  __shared__ float lds_h[EW_][16][32];
  __shared__ int lds_tg[EW_][16];
  __shared__ int lds_sr[EW_][16];

  const int lane = threadIdx.x & 31;
  const int w = threadIdx.x >> 5;
  const int gw = blockIdx.x * EW_ + w;
  const int b = gw / (E_ / 16);
  const int e0 = (gw % (E_ / 16)) * 16;

  // ---- gather phase (lanes 0..15, one edge each)
  if (lane < 16) {
    int e = e0 + lane;
    int s = eidx[e];
    int t = eidx[E_ + e];
    lds_sr[w][lane] = s;
    lds_tg[w][lane] = t;
    const float* xs = x + ((size_t)b * N_ + s) * IN_;
    const float* xt = x + ((size_t)b * N_ + t) * IN_;
    float v7 = 0.f, v8 = 0.f;
#pragma unroll
    for (int k = 0; k < IN_; ++k) {
      float vs = xs[k];
      lds_in[w][lane][k] = vs;
      lds_in[w][lane][IN_ + k] = xt[k];
      if (k == 7) v7 = vs;
      if (k == 8) v8 = vs;
    }
    float ean0 = ec[4 * e + 0], ean1 = ec[4 * e + 1];
    float dir = ec[4 * e + 2], coef = ec[4 * e + 3];
    float ws0 = fmaxf(wstd[0], 1e-6f), ws1 = fmaxf(wstd[1], 1e-6f);
    float spd = fmaxf(v7 * ws0 + wmean[0], 0.0f);
    float wdir = (v8 * ws1 + wmean[1]) * 0.017453292519943295f;  // deg2rad
    float theta = fabsf(dir - wdir);
    float ew = fmaxf(spd * __cosf(theta) * coef, 0.0f);
    lds_in[w][lane][18] = ean0;
    lds_in[w][lane][19] = ean1;
    lds_in[w][lane][20] = ew;
    lds_in[w][lane][21] = 0.f;
    lds_in[w][lane][22] = 0.f;
    lds_in[w][lane][23] = 0.f;
  }
  __syncthreads();

  const int r16 = lane & 15;
  const int koff = (lane < 16) ? 0 : 2;

  // ---- layer 1: (16x21) @ (21x32), K padded to 24 (weights pre-padded)
  v8f c0, c1;
  {
    float b0v = eb1[r16], b1v = eb1[16 + r16];
#pragma unroll
    for (int j = 0; j < 8; ++j) { c0[j] = b0v; c1[j] = b1v; }
  }
#pragma unroll
  for (int ks = 0; ks < 6; ++ks) {
    int k = 4 * ks + koff;
    v2f a;
    a.x = lds_in[w][r16][k];
    a.y = lds_in[w][r16][k + 1];
    v2f b0, b1;
    b0.x = e1p[k * EH_ + r16];
    b0.y = e1p[(k + 1) * EH_ + r16];
    b1.x = e1p[k * EH_ + 16 + r16];
    b1.y = e1p[(k + 1) * EH_ + 16 + r16];
    c0 = wmma4(a, b0, c0);
    c1 = wmma4(a, b1, c1);
  }
  // sigmoid + transpose into A-layout staging tile
#pragma unroll
  for (int j = 0; j < 8; ++j) {
    int r = (lane < 16) ? j : (j + 8);
    lds_h[w][r][r16] = sigf(c0[j]);
    lds_h[w][r][16 + r16] = sigf(c1[j]);
  }
  __syncthreads();

  // ---- layer 2: (16x32) @ (32x30), N padded to 32 (weights pre-padded)
  v8f d0, d1;
  {
    float b0v = eb2p[r16], b1v = eb2p[16 + r16];
#pragma unroll
    for (int j = 0; j < 8; ++j) { d0[j] = b0v; d1[j] = b1v; }
  }
#pragma unroll
  for (int ks = 0; ks < 8; ++ks) {
    int k = 4 * ks + koff;
    v2f a;
    a.x = lds_h[w][r16][k];
    a.y = lds_h[w][r16][k + 1];
    v2f b0, b1;
    b0.x = e2p[k * 32 + r16];
    b0.y = e2p[(k + 1) * 32 + r16];
    b1.x = e2p[k * 32 + 16 + r16];
    b1.y = e2p[(k + 1) * 32 + 16 + r16];
    d0 = wmma4(a, b0, d0);
    d1 = wmma4(a, b1, d1);
  }

  // ---- sigmoid + scatter (+h to tgt, -h to src)
#pragma unroll
  for (int j = 0; j < 8; ++j) {
    int r = (lane < 16) ? j : (j + 8);
    int tg = lds_tg[w][r], sr = lds_sr[w][r];
    size_t baseT = ((size_t)b * N_ + tg) * EO_;
    size_t baseS = ((size_t)b * N_ + sr) * EO_;
    float h0 = sigf(d0[j]);
    atomicAdd(&agg[baseT + r16], h0);
    atomicAdd(&agg[baseS + r16], -h0);
    int c2 = 16 + r16;
    if (c2 < EO_) {
      float h1 = sigf(d1[j]);
      atomicAdd(&agg[baseT + c2], h1);
      atomicAdd(&agg[baseS + c2], -h1);
    }
  }
}

// -------------------------------------------- node layer + xc = [x_gnn, x]
__global__ void k_node(const float* __restrict__ agg, const float* __restrict__ x,
                       const float* __restrict__ nw, const float* __restrict__ nb,
                       float* __restrict__ xc) {
  int tid = blockIdx.x * blockDim.x + threadIdx.x;
  if (tid >= B_ * N_) return;
  float a[EO_];
  const float* ar = agg + (size_t)tid * EO_;
#pragma unroll
  for (int k = 0; k < EO_; ++k) a[k] = ar[k];
  float* out = xc + (size_t)tid * XC_;
#pragma unroll
  for (int c = 0; c < GOUT_; ++c) {
    float s = nb[c];
#pragma unroll
    for (int k = 0; k < EO_; ++k) s += a[k] * nw[k * GOUT_ + c];
    out[c] = sigf(s);
  }
  const float* xr = x + (size_t)tid * IN_;
#pragma unroll
  for (int k = 0; k < IN_; ++k) out[GOUT_ + k] = xr[k];
}

// ---------------------------------------------------------------- GRU kernel
// block = 8 waves covering 2 row-tiles x 4 col-tiles of HID.
// h-tile (32 rows x 64 f32, contiguous 8KB) staged LDS via TDM (TENSORcnt);
// xc-tile staged cooperatively (overlaps the DMA).
__global__ __launch_bounds__(256) void k_gru(
    const float* __restrict__ xc, const float* __restrict__ hin,
    const float* __restrict__ wiT, const float* __restrict__ whT,
    const float* __restrict__ bi, const float* __restrict__ bh,
    float* __restrict__ hout) {
  __shared__ float lds_xc[2][16][24];
  __shared__ float lds_h[2][16][HID_];
  const int tid = threadIdx.x;
  const int rt0 = blockIdx.x * 2;
  const int lane = tid & 31;
  const int w = tid >> 5;

  if (w == 0) {
    // async tensor DMA: 2*16*64 dwords of hin into lds_h
    tdm_load_1d((unsigned)(uintptr_t)&lds_h[0][0][0],
                hin + (size_t)rt0 * 16 * HID_, 2 * 16 * HID_);
  }
  for (int i = tid; i < 2 * 16 * 24; i += 256) {
    int p = i / (16 * 24), rem = i % (16 * 24);
    int r = rem / 24, k = rem % 24;
    int row = (rt0 + p) * 16 + r;
    lds_xc[p][r][k] = (k < XC_) ? xc[(size_t)row * XC_ + k] : 0.0f;
  }
  if (w == 0) __builtin_amdgcn_s_wait_tensorcnt(0);
  __syncthreads();

  const int p = w >> 2;   // row tile within block
  const int ct = w & 3;   // col tile of HID
  const int r16 = lane & 15;
  const int koff = (lane < 16) ? 0 : 2;
  const int col = ct * 16 + r16;  // 0..63

  v8f ir, iz, in_, hr, hz, hn;
  {
    float vir = bi[col], viz = bi[HID_ + col], vin = bi[2 * HID_ + col];
    float vhr = bh[col], vhz = bh[HID_ + col], vhn = bh[2 * HID_ + col];
#pragma unroll
    for (int j = 0; j < 8; ++j) {
      ir[j] = vir; iz[j] = viz; in_[j] = vin;
      hr[j] = vhr; hz[j] = vhz; hn[j] = vhn;
    }
  }

  // gi = xc @ wi.T   (K = 22 padded to 24; wiT pre-padded, loads coalesced)
#pragma unroll
  for (int ks = 0; ks < 6; ++ks) {
    int k = 4 * ks + koff;
    v2f a;
    a.x = lds_xc[p][r16][k];
    a.y = lds_xc[p][r16][k + 1];
    v2f br, bz, bn;
    br.x = wiT[k * G3_ + col];
    br.y = wiT[(k + 1) * G3_ + col];
    bz.x = wiT[k * G3_ + HID_ + col];
    bz.y = wiT[(k + 1) * G3_ + HID_ + col];
    bn.x = wiT[k * G3_ + 2 * HID_ + col];
    bn.y = wiT[(k + 1) * G3_ + 2 * HID_ + col];
    ir = wmma4(a, br, ir);
    iz = wmma4(a, bz, iz);
    in_ = wmma4(a, bn, in_);
  }
  // gh = h @ wh.T    (K = 64)
#pragma unroll
  for (int ks = 0; ks < 16; ++ks) {
    int k = 4 * ks + koff;
    v2f a;
    a.x = lds_h[p][r16][k];
    a.y = lds_h[p][r16][k + 1];
    v2f br, bz, bn;
    br.x = whT[k * G3_ + col];
    br.y = whT[(k + 1) * G3_ + col];
    bz.x = whT[k * G3_ + HID_ + col];
    bz.y = whT[(k + 1) * G3_ + HID_ + col];
    bn.x = whT[k * G3_ + 2 * HID_ + col];
    bn.y = whT[(k + 1) * G3_ + 2 * HID_ + col];
    ir = wmma4(a, br, ir);
    iz = wmma4(a, bz, iz);
    in_ = wmma4(a, bn, in_);
  }

  // gates + state update
#pragma unroll
  for (int j = 0; j < 8; ++j) {
    int r = (lane < 16) ? j : (j + 8);
    float rr = sigf(ir[j] + hr[j]);
    float zz = sigf(iz[j] + hz[j]);
    float nn = tanhf(in_[j] + rr * hn[j]);
    float hold = lds_h[p][r][col];
    int row = (rt0 + p) * 16 + r;
    hout[(size_t)row * HID_ + col] = (1.0f - zz) * nn + zz * hold;
  }
}

// ---------------------------------------------------------------- fc + output
__global__ void k_fc(const float* __restrict__ h, const float* __restrict__ fw,
                     const float* __restrict__ fb, float* __restrict__ xn,
                     float* __restrict__ out, int t) {
  int tid = blockIdx.x * blockDim.x + threadIdx.x;
  if (tid >= B_ * N_) return;
  const float* hr = h + (size_t)tid * HID_;
  float s = fb[0];
#pragma unroll
  for (int k = 0; k < HID_; ++k) s += hr[k] * fw[k];
  xn[tid] = s;
  out[(size_t)tid * PRED_ + t] = s;
}

// ---------------------------------------------------------------- launch
extern "C" void kernel_launch(void* const* d_in, const int* in_sizes, int n_in,
                              void* d_out, int out_size, void* d_ws, size_t ws_size,
                              hipStream_t stream) {
  const float* pm25    = (const float*)d_in[0];
  const float* feature = (const float*)d_in[1];
  const float* eattr   = (const float*)d_in[2];
  const float* wmean   = (const float*)d_in[3];
  const float* wstd    = (const float*)d_in[4];
  const float* ew1     = (const float*)d_in[5];
  const float* eb1     = (const float*)d_in[6];
  const float* ew2     = (const float*)d_in[7];
  const float* eb2     = (const float*)d_in[8];
  const float* nw      = (const float*)d_in[9];
  const float* nb      = (const float*)d_in[10];
  const float* gwi     = (const float*)d_in[11];
  const float* gwh     = (const float*)d_in[12];
  const float* gbi     = (const float*)d_in[13];
  const float* gbh     = (const float*)d_in[14];
  const float* fcw     = (const float*)d_in[15];
  const float* fcb     = (const float*)d_in[16];
  const int*   eidx    = (const int*)d_in[17];
  float* out = (float*)d_out;

  float* ws = (float*)d_ws;
  size_t off = 0;
  float* x    = ws + off; off += (size_t)B_ * N_ * IN_;
  float* agg  = ws + off; off += (size_t)B_ * N_ * EO_;
  float* xc   = ws + off; off += (size_t)B_ * N_ * XC_;
  float* hnA  = ws + off; off += (size_t)B_ * N_ * HID_;
  float* hnB  = ws + off; off += (size_t)B_ * N_ * HID_;
  float* xn   = ws + off; off += (size_t)B_ * N_;
  float* ec   = ws + off; off += (size_t)E_ * 4;
  float* st   = ws + off; off += 16;
  float* wiT  = ws + off; off += 24 * G3_;
  float* whT  = ws + off; off += (size_t)HID_ * G3_;
  float* e1p  = ws + off; off += 24 * EH_;
  float* e2p  = ws + off; off += 32 * 32;
  float* eb2p = ws + off; off += 32;

  // one-time init + edge preprocessing + weight packing
  k_init<<<(B_ * N_ * HID_ + 255) / 256, 256, 0, stream>>>(pm25, hnA, xn);
  k_stats_zero<<<1, 64, 0, stream>>>(st);
  k_stats<<<64, 256, 0, stream>>>(eattr, st);
  k_edge_pre<<<(E_ + 255) / 256, 256, 0, stream>>>(eattr, st, ec);
  k_pack<<<48, 256, 0, stream>>>(gwi, gwh, ew1, ew2, eb2, wiT, whT, e1p, e2p, eb2p);

  for (int t = 0; t < PRED_; ++t) {
    const float* hin = (t & 1) ? hnB : hnA;
    float* hout = (t & 1) ? hnA : hnB;
    k_step_begin<<<(B_ * N_ * EO_ + 255) / 256, 256, 0, stream>>>(agg, x, xn,
                                                                  feature, t);
    k_edge<<<(B_ * (E_ / 16)) / EW_, 256, 0, stream>>>(x, eidx, ec, e1p, eb1,
                                                       e2p, eb2p, wmean, wstd, agg);
    k_node<<<(B_ * N_ + 255) / 256, 256, 0, stream>>>(agg, x, nw, nb, xc);
    k_gru<<<(B_ * N_ / 16) / 2, 256, 0, stream>>>(xc, hin, wiT, whT, gbi, gbh,
                                                  hout);
    k_fc<<<(B_ * N_ + 255) / 256, 256, 0, stream>>>(hout, fcw, fcb, xn, out, t);
  }
}